// DynamicGraphBuilder_57251914056269
// MI455X (gfx1250) — compile-verified
//
#include <hip/hip_runtime.h>
#include <math.h>

#define NROWS 8192
#define DIM   512
#define TOPK  10

typedef __attribute__((ext_vector_type(2))) float v2f;
typedef __attribute__((ext_vector_type(8))) float v8f;

// ---------------------------------------------------------------------------
// Phase 1:  E = X @ W^T + b   (fp32, V_WMMA_F32_16X16X4_F32)
// One wave per TWO 16x16 output tiles (d0 and d0+256), sharing the A fragment.
// A lane m=lane&15 holds X[m][k+klo], X[m][k+klo+1]  (klo = 0 or 2 by half)
// B lane n=lane&15 holds W[n][k+klo], W[n][k+klo+1]  (B = W^T column n)
// ---------------------------------------------------------------------------
__global__ __launch_bounds__(256) void gemm_ew_kernel(
    const float* __restrict__ X,
    const float* __restrict__ Wh, const float* __restrict__ bh,
    const float* __restrict__ Wt, const float* __restrict__ bt,
    float* __restrict__ Eh, float* __restrict__ Et)
{
    const int lane = threadIdx.x & 31;
    const int gwid = blockIdx.x * 8 + (threadIdx.x >> 5);   // global wave id
    // 8192 wave-jobs per matrix: 512 mtiles x 16 dtile-pairs
    const int mat  = gwid >> 13;          // 0: (Wh,bh,Eh)  1: (Wt,bt,Et)
    const int rem  = gwid & 8191;
    const int m0   = (rem >> 4) << 4;     // row-tile * 16
    const int d0   = (rem & 15) << 4;     // first col-tile base; second = d0+256

    const float* W = mat ? Wt : Wh;
    const float* b = mat ? bt : bh;
    float*       E = mat ? Et : Eh;

    const int klo = (lane & 16) ? 2 : 0;
    const float* pa  = X + (m0 + (lane & 15)) * DIM + klo;
    const float* pb0 = W + (d0 + (lane & 15)) * DIM + klo;
    const float* pb1 = pb0 + 256 * DIM;

    v8f acc0 = {}, acc1 = {};
    #pragma unroll 4
    for (int kk = 0; kk < DIM / 4; ++kk) {
        v2f a  = *(const v2f*)(pa  + kk * 4);
        v2f b0 = *(const v2f*)(pb0 + kk * 4);
        v2f b1 = *(const v2f*)(pb1 + kk * 4);
        acc0 = __builtin_amdgcn_wmma_f32_16x16x4_f32(false, a, false, b0,
                                                     (short)0, acc0, false, false);
        acc1 = __builtin_amdgcn_wmma_f32_16x16x4_f32(false, a, false, b1,
                                                     (short)0, acc1, false, false);
    }

    const float bias0 = b[d0 + (lane & 15)];
    const float bias1 = b[d0 + 256 + (lane & 15)];
    const int   mbase = m0 + ((lane & 16) ? 8 : 0);
    #pragma unroll
    for (int v = 0; v < 8; ++v) {
        E[(mbase + v) * DIM + d0 + (lane & 15)]       = acc0[v] + bias0;
        E[(mbase + v) * DIM + d0 + 256 + (lane & 15)] = acc1[v] + bias1;
    }
}

// ---------------------------------------------------------------------------
// Phase 2:  fused  logits = Eh @ Et^T  ->  top-10 / row  ->  softmax -> edges
// Block = 512 threads (16 waves), owns 16 rows. Loops over 16 panels of 512
// columns: WMMA the 16x512 logits panel into LDS (each wave: 2 column tiles
// sharing one A fragment per K-step), then wave w updates the running top-10
// of row w via wave-argmax extraction.
// ---------------------------------------------------------------------------
__global__ __launch_bounds__(512) void topk_kernel(
    const float* __restrict__ Eh, const float* __restrict__ Et,
    int* __restrict__ out_src, int* __restrict__ out_dst,
    float* __restrict__ out_w)
{
    __shared__ float shA[16 * DIM];   // 32 KB: this block's 16 e_h rows
    __shared__ float shL[16 * DIM];   // 32 KB: 16 x 512 logits panel

    const int lane = threadIdx.x & 31;
    const int wv   = threadIdx.x >> 5;        // 0..15, wave wv owns row wv
    const int r0   = blockIdx.x * 16;

    // stage e_h rows into LDS (8192 floats, 2048 float4, 512 threads)
    {
        const float4* src = (const float4*)(Eh + r0 * DIM);
        float4*       dst = (float4*)shA;
        #pragma unroll
        for (int i = 0; i < 4; ++i)
            dst[threadIdx.x + i * 512] = src[threadIdx.x + i * 512];
    }
    __syncthreads();

    // running top-10 (sorted descending), replicated across all lanes of wave
    float tv[TOPK]; int ti[TOPK];
    #pragma unroll
    for (int i = 0; i < TOPK; ++i) { tv[i] = -INFINITY; ti[i] = 0; }

    const int klo = (lane & 16) ? 2 : 0;
    const float* aBase = shA + (lane & 15) * DIM + klo;

    for (int p = 0; p < NROWS / DIM; ++p) {
        const int pb = p * DIM;                        // panel base column

        // ---- GEMM: wave wv computes column tiles wv and wv+16 together ----
        {
            const float* bBase0 = Et + (pb + wv * 16 + (lane & 15)) * DIM + klo;
            const float* bBase1 = bBase0 + 256 * DIM;  // tile wv+16
            v8f acc0 = {}, acc1 = {};
            #pragma unroll 4
            for (int kk = 0; kk < DIM / 4; ++kk) {
                v2f a  = *(const v2f*)(aBase  + kk * 4);   // shared A (LDS)
                v2f b0 = *(const v2f*)(bBase0 + kk * 4);
                v2f b1 = *(const v2f*)(bBase1 + kk * 4);
                acc0 = __builtin_amdgcn_wmma_f32_16x16x4_f32(false, a, false, b0,
                                                             (short)0, acc0, false, false);
                acc1 = __builtin_amdgcn_wmma_f32_16x16x4_f32(false, a, false, b1,
                                                             (short)0, acc1, false, false);
            }
            const int cb0   = wv * 16 + (lane & 15);
            const int rbase = (lane & 16) ? 8 : 0;
            #pragma unroll
            for (int v = 0; v < 8; ++v) {
                shL[(rbase + v) * DIM + cb0]       = acc0[v];
                shL[(rbase + v) * DIM + cb0 + 256] = acc1[v];
            }
        }
        __syncthreads();

        // ---- top-k update for row wv over this 512-wide panel ----
        float val[16];
        #pragma unroll
        for (int t = 0; t < 16; ++t) val[t] = shL[wv * DIM + t * 32 + lane];

        for (;;) {
            // lane-local argmax (lower t wins ties -> lower global index)
            float lv = val[0]; int lt = 0;
            #pragma unroll
            for (int t = 1; t < 16; ++t)
                if (val[t] > lv) { lv = val[t]; lt = t; }
            float mv = lv;
            int   mi = pb + lt * 32 + lane;
            // wave argmax, lower index wins ties (matches jax top_k)
            #pragma unroll
            for (int off = 16; off >= 1; off >>= 1) {
                float ov = __shfl_xor(mv, off, 32);
                int   oi = __shfl_xor(mi, off, 32);
                if (ov > mv || (ov == mv && oi < mi)) { mv = ov; mi = oi; }
            }
            if (!(mv > tv[TOPK - 1])) break;           // uniform across wave

            // replicated sorted insert
            float cv = mv; int ci = mi;
            #pragma unroll
            for (int i = 0; i < TOPK; ++i) {
                if (cv > tv[i]) {
                    float t0 = tv[i]; int t1 = ti[i];
                    tv[i] = cv; ti[i] = ci; cv = t0; ci = t1;
                }
            }
            // owner lane removes the extracted element
            const int d = mi - pb;
            if ((d & 31) == lane) {
                const int tw = d >> 5;
                #pragma unroll
                for (int t = 0; t < 16; ++t)
                    if (t == tw) val[t] = -INFINITY;
            }
        }
        __syncthreads();   // shL reused next panel
    }

    // ---- softmax over the 10 kept logits, emit edges ----
    const float scale = 0.044194173824159216f;         // 512 ** -0.5
    const float m = tv[0] * scale;
    float sum = 0.f;
    #pragma unroll
    for (int i = 0; i < TOPK; ++i) sum += expf(tv[i] * scale - m);

    // lane i needs element i of the replicated lists (select chain, no scratch)
    float myv = tv[0]; int myi = ti[0];
    #pragma unroll
    for (int i = 1; i < TOPK; ++i)
        if (lane == i) { myv = tv[i]; myi = ti[i]; }

    const int row = r0 + wv;
    if (lane < TOPK) {
        const int o = row * TOPK + lane;
        out_src[o] = row;
        out_dst[o] = myi;
        out_w[o]   = expf(myv * scale - m) / sum;
    }
}

// ---------------------------------------------------------------------------
extern "C" void kernel_launch(void* const* d_in, const int* in_sizes, int n_in,
                              void* d_out, int out_size, void* d_ws, size_t ws_size,
                              hipStream_t stream) {
    const float* X  = (const float*)d_in[0];
    const float* Wh = (const float*)d_in[1];
    const float* bh = (const float*)d_in[2];
    const float* Wt = (const float*)d_in[3];
    const float* bt = (const float*)d_in[4];

    float* Eh = (float*)d_ws;               // 8192*512 fp32 = 16 MB
    float* Et = Eh + NROWS * DIM;           // 16 MB

    int*   out_src = (int*)d_out;                       // edge_index[0]
    int*   out_dst = out_src + NROWS * TOPK;            // edge_index[1]
    float* out_w   = (float*)d_out + 2 * NROWS * TOPK;  // edge_weight

    // 2 matrices * 512 mtiles * 16 dtile-pairs = 16384 waves; 8 waves / block
    gemm_ew_kernel<<<2048, 256, 0, stream>>>(X, Wh, bh, Wt, bt, Eh, Et);
    topk_kernel<<<NROWS / 16, 512, 0, stream>>>(Eh, Et, out_src, out_dst, out_w);
}